// TripletLossAlignedReID_54589034332824
// MI455X (gfx1250) — compile-verified
//
#include <hip/hip_runtime.h>
#include <hip/hip_bf16.h>
#include <math.h>

typedef __attribute__((ext_vector_type(2))) float v2f;
typedef __attribute__((ext_vector_type(8))) float v8f;

#define N_SAMP 512
#define DIM    2048
#define LDIM   128
#define PARTS  8
#define MARGIN 0.3f
#define BIGF   1e9f

// ---------------------------------------------------------------------------
// Kernel 1: row squared norms  sq[i] = sum_k X[i][k]^2      grid=512, block=32
// ---------------------------------------------------------------------------
__global__ void row_norms_kernel(const float* __restrict__ X, float* __restrict__ sq) {
    const int row  = blockIdx.x;
    const int lane = threadIdx.x;              // 0..31 (one wave)
    const float* p = X + (size_t)row * DIM;
    float s = 0.0f;
    for (int k = lane; k < DIM; k += 32) {
        float v = p[k];
        s += v * v;
    }
    for (int off = 16; off > 0; off >>= 1)
        s += __shfl_down(s, off);
    if (lane == 0) sq[row] = s;
}

// ---------------------------------------------------------------------------
// Kernel 2: pairwise distances via WMMA f32 16x16x4, 2x2 register blocking.
// One wave computes a 32x32 block of G = X * X^T (4 WMMA tiles), reusing each
// A/B fragment twice -> 1 global_load_b64 per WMMA. Then
//   dist[i][j] = sqrt(max(sq[i]+sq[j]-2*G[i][j], 1e-12)).
// grid = 64 blocks * 4 waves = 256 super-tiles (16x16 grid of 32x32 blocks).
// ---------------------------------------------------------------------------
__global__ void pairwise_dist_wmma_kernel(const float* __restrict__ X,
                                          const float* __restrict__ sq,
                                          float* __restrict__ dist) {
    const int wave = threadIdx.x >> 5;         // 0..3
    const int lane = threadIdx.x & 31;
    const int st = blockIdx.x * 4 + wave;      // 0..255
    const int sm = st >> 4;                    // row super-tile 0..15
    const int sn = st & 15;                    // col super-tile 0..15

    const int half = lane >> 4;                // 0: K={0,1}, 1: K={2,3}
    const int idx  = lane & 15;                // matrix row within 16-tile

    // A 16x4 f32 layout: lane(half,idx) holds A[idx][k+2*half .. +1] (2 VGPRs).
    // B 4x16 f32 layout: VGPR r = row K=r (lanes 0-15) / K=r+2 (lanes 16-31),
    // so lane(half,idx) holds B[k+2*half .. +1][idx] = Xcol[idx][k+2*half .. +1].
    const float* a0 = X + (size_t)(sm * 32 + idx) * DIM;
    const float* a1 = a0 + (size_t)16 * DIM;
    const float* b0 = X + (size_t)(sn * 32 + idx) * DIM;
    const float* b1 = b0 + (size_t)16 * DIM;

    v8f c00 = {}, c01 = {}, c10 = {}, c11 = {};
#pragma unroll 4
    for (int k = 0; k < DIM; k += 4) {
        const int ka = k + half * 2;
        v2f ra0 = *(const v2f*)(a0 + ka);
        v2f ra1 = *(const v2f*)(a1 + ka);
        v2f rb0 = *(const v2f*)(b0 + ka);
        v2f rb1 = *(const v2f*)(b1 + ka);
        c00 = __builtin_amdgcn_wmma_f32_16x16x4_f32(false, ra0, false, rb0,
                                                    (short)0, c00, false, false);
        c01 = __builtin_amdgcn_wmma_f32_16x16x4_f32(false, ra0, false, rb1,
                                                    (short)0, c01, false, false);
        c10 = __builtin_amdgcn_wmma_f32_16x16x4_f32(false, ra1, false, rb0,
                                                    (short)0, c10, false, false);
        c11 = __builtin_amdgcn_wmma_f32_16x16x4_f32(false, ra1, false, rb1,
                                                    (short)0, c11, false, false);
    }

    // C/D layout: VGPR r -> M = r + 8*half, N = idx.
    const int j0 = sn * 32 + idx;
    const int j1 = j0 + 16;
    const float sqj0 = sq[j0];
    const float sqj1 = sq[j1];
#pragma unroll
    for (int r = 0; r < 8; ++r) {
        const int i0 = sm * 32 + r + 8 * half;
        const int i1 = i0 + 16;
        const float sqi0 = sq[i0];
        const float sqi1 = sq[i1];
        dist[(size_t)i0 * N_SAMP + j0] = sqrtf(fmaxf(sqi0 + sqj0 - 2.0f * c00[r], 1e-12f));
        dist[(size_t)i0 * N_SAMP + j1] = sqrtf(fmaxf(sqi0 + sqj1 - 2.0f * c01[r], 1e-12f));
        dist[(size_t)i1 * N_SAMP + j0] = sqrtf(fmaxf(sqi1 + sqj0 - 2.0f * c10[r], 1e-12f));
        dist[(size_t)i1 * N_SAMP + j1] = sqrtf(fmaxf(sqi1 + sqj1 - 2.0f * c11[r], 1e-12f));
    }
}

// ---------------------------------------------------------------------------
// Kernel 3: hard mining. One wave per row: max over positives (argmax),
// min over negatives (argmin).                              grid=512, block=32
// ---------------------------------------------------------------------------
__global__ void hard_mine_kernel(const float* __restrict__ dist,
                                 const int* __restrict__ targets,
                                 float* __restrict__ dist_ap, float* __restrict__ dist_an,
                                 int* __restrict__ p_inds, int* __restrict__ n_inds) {
    const int row  = blockIdx.x;
    const int lane = threadIdx.x;
    const int t    = targets[row];
    const float* drow = dist + (size_t)row * N_SAMP;

    float best_ap = -BIGF, best_an = BIGF;
    int   bpi = 0,         bni = 0;
    for (int j = lane; j < N_SAMP; j += 32) {
        const float d = drow[j];
        if (targets[j] == t) {
            if (d > best_ap) { best_ap = d; bpi = j; }
        } else {
            if (d < best_an) { best_an = d; bni = j; }
        }
    }
    for (int off = 16; off > 0; off >>= 1) {
        float oap = __shfl_down(best_ap, off);
        int   opi = __shfl_down(bpi, off);
        float oan = __shfl_down(best_an, off);
        int   oni = __shfl_down(bni, off);
        if (oap > best_ap) { best_ap = oap; bpi = opi; }
        if (oan < best_an) { best_an = oan; bni = oni; }
    }
    if (lane == 0) {
        dist_ap[row] = best_ap;
        dist_an[row] = best_an;
        p_inds[row]  = bpi;
        n_inds[row]  = bni;
    }
}

// ---------------------------------------------------------------------------
// Kernel 4: local aligned distance. Block of 64 threads per sample:
// stage lf[i], lf[p], lf[n] (layout [d][part], contiguous 1024 floats) in LDS,
// each thread computes one (p,q) pair of the 8x8 matrices for ap and an,
// tanh(d/2), then DTW shortest path.                        grid=512, block=64
// ---------------------------------------------------------------------------
__global__ void local_dist_kernel(const float* __restrict__ lf,
                                  const int* __restrict__ p_inds,
                                  const int* __restrict__ n_inds,
                                  float* __restrict__ local_ap,
                                  float* __restrict__ local_an) {
    __shared__ float sA[LDIM * PARTS];
    __shared__ float sP[LDIM * PARTS];
    __shared__ float sN[LDIM * PARTS];
    __shared__ float sDp[64];
    __shared__ float sDn[64];

    const int i   = blockIdx.x;
    const int tid = threadIdx.x;               // 0..63
    const float* a  = lf + (size_t)i * (LDIM * PARTS);
    const float* bp = lf + (size_t)p_inds[i] * (LDIM * PARTS);
    const float* bn = lf + (size_t)n_inds[i] * (LDIM * PARTS);

    for (int k = tid; k < LDIM * PARTS; k += 64) {
        sA[k] = a[k];
        sP[k] = bp[k];
        sN[k] = bn[k];
    }
    __syncthreads();

    // lf element (sample, part p, dim d) lives at [d*PARTS + p].
    const int p = tid >> 3;                    // row of 8x8 matrix
    const int q = tid & 7;                     // col
    float s_p = 0.0f, s_n = 0.0f;
#pragma unroll 4
    for (int d = 0; d < LDIM; ++d) {
        const float av  = sA[d * PARTS + p];
        const float bpv = sP[d * PARTS + q];
        const float bnv = sN[d * PARTS + q];
        const float ep = av - bpv;
        const float en = av - bnv;
        s_p += ep * ep;
        s_n += en * en;
    }
    sDp[tid] = tanhf(0.5f * sqrtf(fmaxf(s_p, 1e-12f)));
    sDn[tid] = tanhf(0.5f * sqrtf(fmaxf(s_n, 1e-12f)));
    __syncthreads();

    // DTW: dp[r][c] = D[r][c] + min(dp[r-1][c], dp[r][c-1]); 8x8, serial (tiny).
    if (tid < 2) {
        const float* D = (tid == 0) ? sDp : sDn;
        float dp[PARTS];
        float run = 0.0f;
        for (int cc = 0; cc < PARTS; ++cc) { run += D[cc]; dp[cc] = run; }
        for (int r = 1; r < PARTS; ++r) {
            dp[0] += D[r * PARTS];
            for (int cc = 1; cc < PARTS; ++cc)
                dp[cc] = D[r * PARTS + cc] + fminf(dp[cc], dp[cc - 1]);
        }
        if (tid == 0) local_ap[i] = dp[PARTS - 1];
        else          local_an[i] = dp[PARTS - 1];
    }
}

// ---------------------------------------------------------------------------
// Kernel 5: margin ranking loss mean(relu(margin + ap - an)). 1 block of 256.
// ---------------------------------------------------------------------------
__global__ void margin_mean_kernel(const float* __restrict__ ap,
                                   const float* __restrict__ an,
                                   float* __restrict__ out, int out_idx) {
    __shared__ float buf[8];
    const int tid = threadIdx.x;               // 0..255
    float s = 0.0f;
    for (int i = tid; i < N_SAMP; i += 256)
        s += fmaxf(0.0f, MARGIN + ap[i] - an[i]);
    for (int off = 16; off > 0; off >>= 1)
        s += __shfl_down(s, off);
    if ((tid & 31) == 0) buf[tid >> 5] = s;
    __syncthreads();
    if (tid == 0) {
        float total = 0.0f;
        for (int w = 0; w < 8; ++w) total += buf[w];
        out[out_idx] = total / (float)N_SAMP;
    }
}

// ---------------------------------------------------------------------------
extern "C" void kernel_launch(void* const* d_in, const int* in_sizes, int n_in,
                              void* d_out, int out_size, void* d_ws, size_t ws_size,
                              hipStream_t stream) {
    const float* X       = (const float*)d_in[0];   // [512, 2048]
    const int*   targets = (const int*)d_in[1];     // [512]
    const float* lf      = (const float*)d_in[2];   // [512, 128, 8]
    float* out = (float*)d_out;                     // [global_loss, local_loss]

    // workspace layout (floats / ints)
    float* dist    = (float*)d_ws;                  // 512*512
    float* sq      = dist + N_SAMP * N_SAMP;        // 512
    float* dist_ap = sq + N_SAMP;                   // 512
    float* dist_an = dist_ap + N_SAMP;              // 512
    float* loc_ap  = dist_an + N_SAMP;              // 512
    float* loc_an  = loc_ap + N_SAMP;               // 512
    int*   p_inds  = (int*)(loc_an + N_SAMP);       // 512
    int*   n_inds  = p_inds + N_SAMP;               // 512

    row_norms_kernel<<<N_SAMP, 32, 0, stream>>>(X, sq);
    pairwise_dist_wmma_kernel<<<64, 128, 0, stream>>>(X, sq, dist);
    hard_mine_kernel<<<N_SAMP, 32, 0, stream>>>(dist, targets, dist_ap, dist_an,
                                                p_inds, n_inds);
    margin_mean_kernel<<<1, 256, 0, stream>>>(dist_ap, dist_an, out, 0);
    local_dist_kernel<<<N_SAMP, 64, 0, stream>>>(lf, p_inds, n_inds, loc_ap, loc_an);
    margin_mean_kernel<<<1, 256, 0, stream>>>(loc_ap, loc_an, out, 1);
}